// MultiHeadAttention_11132555231520
// MI455X (gfx1250) — compile-verified
//
#include <hip/hip_runtime.h>
#include <stddef.h>
#include <stdint.h>

// ---------------------------------------------------------------------------
// Problem constants (from the reference)
// ---------------------------------------------------------------------------
#define BATCH   2
#define SEQ     2048
#define HID     1024
#define NHEADS  16
#define DHEAD   64
#define ROWS    (BATCH * SEQ)        // 4096
#define LN_EPS  1e-5f

typedef float v2f __attribute__((ext_vector_type(2)));
typedef float v8f __attribute__((ext_vector_type(8)));
typedef unsigned int uint4v __attribute__((ext_vector_type(4)));
typedef int int8v __attribute__((ext_vector_type(8)));

// D(16x16,f32) = A(16x4,f32) * B(4x16,f32) + C   -- CDNA5 wave32 WMMA
__device__ __forceinline__ v8f wmma_f32(v2f a, v2f b, v8f c) {
    return __builtin_amdgcn_wmma_f32_16x16x4_f32(
        /*neg_a=*/false, a, /*neg_b=*/false, b,
        /*c_mod=*/(short)0, c, /*reuse_a=*/false, /*reuse_b=*/false);
}

// ---------------------------------------------------------------------------
// Tensor Data Mover: DMA a 64-row x 32-dword fp32 tile (row stride K dwords in
// memory) into LDS, writing 32 data dwords + 4 pad dwords per row (stride 36
// floats -> 16B-aligned rows, conflict-free column reads).
// D# per CDNA5 ISA. Issued once per wave; EXEC ignored.
// ---------------------------------------------------------------------------
#define TM 64
#define TN 64
#define KC 32
#define KP 36                        // padded LDS row stride (floats)

__device__ __forceinline__ void tdm_load_tile(const float* gsrc,
                                              uint32_t lds_off,
                                              int nrows, int K)
{
    const uint64_t ga = (uint64_t)(uintptr_t)gsrc;
    uint4v g0;
    g0.x = 1u;                                   // count=1, user desc
    g0.y = lds_off;                              // lds_addr (bytes)
    g0.z = (uint32_t)ga;                         // global_addr[31:0]
    g0.w = (uint32_t)(ga >> 32) | 0x80000000u;   // global_addr[56:32] | type=2

    int8v g1;
    // data_size=2 (4B) | pad_enable | pad_interval=4 (32 dwords) | pad_amount=3 (4 dwords)
    g1[0] = 0x07120000;
    g1[1] = (int)((K & 0xFFFF) << 16);                           // tensor_dim0[15:0]
    g1[2] = (int)(((K >> 16) & 0xFFFF) | ((nrows & 0xFFFF) << 16)); // dim0 hi | dim1 lo
    g1[3] = (int)(((nrows >> 16) & 0xFFFF) | (KC << 16));        // dim1 hi | tile_dim0=32
    g1[4] = TM;                                  // tile_dim1 = 64 rows (tile_dim2=0)
    g1[5] = K;                                   // tensor_dim0_stride[31:0]
    g1[6] = 0;
    g1[7] = 0;

    // 2D tensor: VADDR2/VADDR3 omitted (NULL)
    asm volatile("tensor_load_to_lds %0, %1" :: "s"(g0), "s"(g1) : "memory");
}

// ---------------------------------------------------------------------------
// GEMM:  C[M][N] = X[M][K] @ W[N][K]^T     (nn.Linear, bias=false)
// 256 threads = 8 waves, 64x64 tile, TDM double-buffered K-chunks of 32.
// Wave (wy in 0..3, wx in 0..1) owns a 16x32 sub-tile (two 16x16 WMMA accs).
// ---------------------------------------------------------------------------
__global__ __launch_bounds__(256)
void gemm_xwT(const float* __restrict__ X, const float* __restrict__ W,
              float* __restrict__ C, int M, int N, int K)
{
    __shared__ float Xs[2][TM * KP];
    __shared__ float Ws[2][TN * KP];

    const int tid  = threadIdx.x;
    const int wave = tid >> 5;
    const int lane = tid & 31;
    const int half = lane >> 4;        // 0: lanes 0-15, 1: lanes 16-31
    const int lm   = lane & 15;
    const int wy   = wave & 3;         // 4 row-groups of 16
    const int wx   = wave >> 2;        // 2 col-groups of 32
    const int bm   = blockIdx.x * TM;
    const int bn   = blockIdx.y * TN;

    v8f acc0 = {};
    v8f acc1 = {};

    if (wave == 0) {                   // prefetch chunk 0
        tdm_load_tile(X + (size_t)bm * K, (uint32_t)(uintptr_t)&Xs[0][0], M, K);
        tdm_load_tile(W + (size_t)bn * K, (uint32_t)(uintptr_t)&Ws[0][0], N, K);
    }

    const int nchunk = K / KC;
    int buf = 0;
    for (int i = 0; i < nchunk; ++i) {
        if (wave == 0)
            asm volatile("s_wait_tensorcnt 0x0" ::: "memory");
        __syncthreads();               // chunk i landed; buf^1 free (reads done)

        if (wave == 0 && (i + 1) < nchunk) {
            const int k0 = (i + 1) * KC;
            tdm_load_tile(X + (size_t)bm * K + k0,
                          (uint32_t)(uintptr_t)&Xs[buf ^ 1][0], M, K);
            tdm_load_tile(W + (size_t)bn * K + k0,
                          (uint32_t)(uintptr_t)&Ws[buf ^ 1][0], N, K);
        }

        const float* xs = &Xs[buf][0];
        const float* ws = &Ws[buf][0];
        #pragma unroll
        for (int k = 0; k < KC; k += 4) {
            // A frag: m = lm, kk = k + 2*half + v
            v2f a, b0, b1;
            a.x  = xs[(wy * 16 + lm) * KP + k + 2 * half];
            a.y  = xs[(wy * 16 + lm) * KP + k + 2 * half + 1];
            // B frags: B[kk][n] = W[n][kk]
            b0.x = ws[(wx * 32 + lm) * KP + k + 2 * half];
            b0.y = ws[(wx * 32 + lm) * KP + k + 2 * half + 1];
            b1.x = ws[(wx * 32 + 16 + lm) * KP + k + 2 * half];
            b1.y = ws[(wx * 32 + 16 + lm) * KP + k + 2 * half + 1];
            acc0 = wmma_f32(a, b0, acc0);
            acc1 = wmma_f32(a, b1, acc1);
        }
        buf ^= 1;
    }

    // D layout: VGPR g, lanes 0-15 -> M=g, lanes 16-31 -> M=g+8; N = lm.
    const int row0 = bm + wy * 16;
    const int col0 = bn + wx * 32;
    #pragma unroll
    for (int g = 0; g < 8; ++g) {
        int m = g + 8 * half;
        C[(size_t)(row0 + m) * N + col0 + lm]      = acc0[g];
        C[(size_t)(row0 + m) * N + col0 + 16 + lm] = acc1[g];
    }
}

// ---------------------------------------------------------------------------
// Fused attention for one (b, h, 16-query block):
//   Q tile staged via global_load_async_to_lds_b128 (ASYNCcnt path),
//   scores (WMMA, LDS-resident 16x2048 slab) -> masked softmax ->
//   attn written NT to global -> context = P @ V (WMMA) -> ctx buffer.
// 256 threads = 8 waves.  Deterministic (no float atomics).
// ---------------------------------------------------------------------------
#define QB    16
#define SPAD  2049                    // score row stride (2048 % 64 == 0 fix)
#define QPAD  68                      // Q row stride: 16B aligned, (4m+k)%64 distinct

__global__ __launch_bounds__(256)
void attn_kernel(const float* __restrict__ Qg, const float* __restrict__ Kg,
                 const float* __restrict__ Vg, const int* __restrict__ mask,
                 float* __restrict__ attn_out, float* __restrict__ ctx_out)
{
    extern __shared__ float smem[];
    float* Ss   = smem;                       // [QB][SPAD]   scores / probs
    float* Qs   = Ss  + QB * SPAD;            // [QB][QPAD]   Q tile
    float* Cw   = Qs  + QB * QPAD;            // [8][QB][64]  per-wave context
    float* red  = Cw  + 8 * QB * 64;          // [16][16]
    float* rmax = red + 256;                  // [16]
    float* rsum = rmax + 16;                  // [16]

    const int tid  = threadIdx.x;
    const int wave = tid >> 5;
    const int lane = tid & 31;
    const int half = lane >> 4;
    const int lm   = lane & 15;

    const int nqb = SEQ / QB;                 // 128
    const int qb  = blockIdx.x % nqb;
    const int h   = (blockIdx.x / nqb) % NHEADS;
    const int b   = blockIdx.x / (nqb * NHEADS);
    const int q0  = qb * QB;

    // ---- stage Q tile (16x64) asynchronously: one b128 per thread ----
    {
        const int r  = tid >> 4;              // row 0..15
        const int f4 = tid & 15;              // float4 index within row
        const uint32_t ldsa =
            (uint32_t)(uintptr_t)&Qs[r * QPAD + f4 * 4];
        const uint64_t ga = (uint64_t)(uintptr_t)
            (Qg + (size_t)(b * SEQ + q0 + r) * HID + h * DHEAD + f4 * 4);
        asm volatile("global_load_async_to_lds_b128 %0, %1, off"
                     :: "v"(ldsa), "v"(ga) : "memory");
        asm volatile("s_wait_asynccnt 0x0" ::: "memory");
    }
    __syncthreads();

    // ---- phase 1: scores = (Q @ K^T) * 1/8, key tiles striped over waves ----
    for (int kt = wave; kt < nqb; kt += 8) {
        const float* kb = Kg + (size_t)(b * SEQ + kt * QB + lm) * HID + h * DHEAD;
        v8f acc = {};
        #pragma unroll
        for (int k = 0; k < DHEAD; k += 4) {
            v2f a, bb;
            a.x  = Qs[lm * QPAD + k + 2 * half];
            a.y  = Qs[lm * QPAD + k + 2 * half + 1];
            bb.x = kb[k + 2 * half];          // B[kk][n] = Key[n][kk]
            bb.y = kb[k + 2 * half + 1];
            acc = wmma_f32(a, bb, acc);
        }
        #pragma unroll
        for (int g = 0; g < 8; ++g) {
            int m = g + 8 * half;
            Ss[m * SPAD + kt * QB + lm] = acc[g] * 0.125f;
        }
    }
    __syncthreads();

    // ---- phase 2: masked softmax over each of the 16 rows ----
    const int row = tid >> 4;                 // 16 threads per row
    const int t   = tid & 15;
    const size_t mrow = ((size_t)(b * SEQ + q0 + row)) * SEQ;

    float mx = -3.402823466e38f;
    for (int i = 0; i < SEQ / 16; ++i) {
        int c = i * 16 + t;                   // coalesced mask reads
        float v = Ss[row * SPAD + c];
        v = (mask[mrow + c] != 0) ? v : -1e9f;
        Ss[row * SPAD + c] = v;
        mx = fmaxf(mx, v);
    }
    red[row * 16 + t] = mx;
    __syncthreads();
    if (t == 0) {
        float m = red[row * 16];
        for (int j = 1; j < 16; ++j) m = fmaxf(m, red[row * 16 + j]);
        rmax[row] = m;
    }
    __syncthreads();

    const float rm = rmax[row];
    float sum = 0.0f;
    for (int i = 0; i < SEQ / 16; ++i) {
        int c = i * 16 + t;
        float e = __expf(Ss[row * SPAD + c] - rm);
        Ss[row * SPAD + c] = e;               // keep unnormalized exp in LDS
        sum += e;
    }
    red[row * 16 + t] = sum;
    __syncthreads();
    if (t == 0) {
        float s = 0.0f;
        for (int j = 0; j < 16; ++j) s += red[row * 16 + j];
        rsum[row] = 1.0f / s;
    }
    __syncthreads();

    // stream normalized attn row out (write-once, non-temporal)
    {
        const float inv = rsum[row];
        const size_t abase =
            ((size_t)((b * NHEADS + h) * SEQ + q0 + row)) * (size_t)SEQ;
        for (int i = 0; i < SEQ / 16; ++i) {
            int c = i * 16 + t;
            __builtin_nontemporal_store(Ss[row * SPAD + c] * inv,
                                        &attn_out[abase + c]);
        }
    }
    __syncthreads();

    // ---- phase 3: context = P @ V, K-range striped over waves ----
    {
        v8f acc[4] = {{}, {}, {}, {}};
        const int kb0 = wave * (SEQ / 8);     // 256 keys per wave
        for (int k = kb0; k < kb0 + SEQ / 8; k += 4) {
            v2f a;
            a.x = Ss[lm * SPAD + k + 2 * half];
            a.y = Ss[lm * SPAD + k + 2 * half + 1];
            const int r0 = k + 2 * half;
            const float* v0 = Vg + (size_t)(b * SEQ + r0) * HID + h * DHEAD;
            const float* v1 = v0 + HID;
            #pragma unroll
            for (int dt = 0; dt < 4; ++dt) {
                v2f bb;
                bb.x = v0[dt * 16 + lm];      // B[kk][d] = V[kk][d]
                bb.y = v1[dt * 16 + lm];
                acc[dt] = wmma_f32(a, bb, acc[dt]);
            }
        }
        // per-wave private partial tile (deterministic reduction later)
        float* cw = Cw + wave * (QB * 64);
        #pragma unroll
        for (int dt = 0; dt < 4; ++dt)
            #pragma unroll
            for (int g = 0; g < 8; ++g) {
                int m = g + 8 * half;
                cw[m * 64 + dt * 16 + lm] = acc[dt][g];
            }
    }
    __syncthreads();

    // reduce the 8 per-wave tiles in fixed order, normalize, write ctx
    #pragma unroll
    for (int i = 0; i < 4; ++i) {
        int e = tid + 256 * i;                // 1024 elements
        int r = e >> 6, d = e & 63;
        float s = 0.0f;
        #pragma unroll
        for (int w = 0; w < 8; ++w) s += Cw[w * (QB * 64) + r * 64 + d];
        ctx_out[(size_t)(b * SEQ + q0 + r) * HID + h * DHEAD + d] =
            s * rsum[r];
    }
}

// ---------------------------------------------------------------------------
// out = LayerNorm(proj + resid), one 256-thread workgroup per row of 1024.
// ---------------------------------------------------------------------------
__global__ __launch_bounds__(256)
void add_ln_kernel(const float* __restrict__ proj,
                   const float* __restrict__ resid,
                   float* __restrict__ out)
{
    __shared__ float sred[256];
    const int row = blockIdx.x;
    const int tid = threadIdx.x;
    const float* p = proj  + (size_t)row * HID;
    const float* r = resid + (size_t)row * HID;

    float v[4];
    float s = 0.0f;
    #pragma unroll
    for (int i = 0; i < 4; ++i) {
        v[i] = p[tid + 256 * i] + r[tid + 256 * i];
        s += v[i];
    }
    sred[tid] = s;
    __syncthreads();
    for (int off = 128; off > 0; off >>= 1) {
        if (tid < off) sred[tid] += sred[tid + off];
        __syncthreads();
    }
    const float mu = sred[0] * (1.0f / HID);
    __syncthreads();

    float vs = 0.0f;
    #pragma unroll
    for (int i = 0; i < 4; ++i) {
        float d = v[i] - mu;
        vs += d * d;
    }
    sred[tid] = vs;
    __syncthreads();
    for (int off = 128; off > 0; off >>= 1) {
        if (tid < off) sred[tid] += sred[tid + off];
        __syncthreads();
    }
    const float inv = rsqrtf(sred[0] * (1.0f / HID) + LN_EPS);

    #pragma unroll
    for (int i = 0; i < 4; ++i)
        out[(size_t)row * HID + tid + 256 * i] = (v[i] - mu) * inv;
}

// ---------------------------------------------------------------------------
// launch
// ---------------------------------------------------------------------------
extern "C" void kernel_launch(void* const* d_in, const int* in_sizes, int n_in,
                              void* d_out, int out_size, void* d_ws, size_t ws_size,
                              hipStream_t stream)
{
    (void)in_sizes; (void)n_in; (void)out_size; (void)ws_size;

    const float* X    = (const float*)d_in[0];   // [B,S,HID]
    const int*   mask = (const int*)  d_in[1];   // [B,S,S]
    const float* WQ   = (const float*)d_in[2];
    const float* WK   = (const float*)d_in[3];
    const float* WV   = (const float*)d_in[4];
    const float* WO   = (const float*)d_in[5];

    float* out  = (float*)d_out;                          // [B,S,HID]
    float* attn = out + (size_t)ROWS * HID;               // [B,H,S,S]

    const size_t mat = (size_t)ROWS * HID;                // 4M floats
    float* Q    = (float*)d_ws;
    float* K    = Q    + mat;
    float* V    = K    + mat;
    float* ctx  = V    + mat;
    float* proj = ctx  + mat;

    dim3 gg(ROWS / TM, HID / TN);                         // 64 x 16

    gemm_xwT<<<gg, 256, 0, stream>>>(X, WQ, Q, ROWS, HID, HID);
    gemm_xwT<<<gg, 256, 0, stream>>>(X, WK, K, ROWS, HID, HID);
    gemm_xwT<<<gg, 256, 0, stream>>>(X, WV, V, ROWS, HID, HID);

    const size_t smem_attn =
        (size_t)(QB * SPAD + QB * QPAD + 8 * QB * 64 + 256 + 16 + 16) *
        sizeof(float);                                    // ~169 KB of 320 KB LDS
    attn_kernel<<<BATCH * NHEADS * (SEQ / QB), 256, smem_attn, stream>>>(
        Q, K, V, mask, attn, ctx);

    gemm_xwT<<<gg, 256, 0, stream>>>(ctx, WO, proj, ROWS, HID, HID);
    add_ln_kernel<<<ROWS, 256, 0, stream>>>(proj, X, out);
}